// AEVComputer_76063870812526
// MI455X (gfx1250) — compile-verified
//
#include <hip/hip_runtime.h>
#include <math.h>

typedef float v2f __attribute__((ext_vector_type(2)));
typedef float v4f __attribute__((ext_vector_type(4)));
typedef float v8f __attribute__((ext_vector_type(8)));

#define ATOMS   48
#define NSLOT   47          // neighbors per center
#define NTRI    1081        // C(47,2)
#define NITER   34          // ceil(NTRI/32)
#define NWAVE   4           // waves per block (wave32)
#define FEAT    384         // 4*16 radial + 10*32 angular
#define PI_F    3.14159265358979323846f
#define RCR_F   5.2f
#define RCA_F   3.5f

// One 128-thread block (4 wave32s) per (molecule, center-atom).
// Binned scatter-adds are one-hot segment-sum matmuls with
// V_WMMA_F32_16X16X4_F32: D[bin, feat] += onehot[bin, triple] x at[triple, feat].
// Each wave owns a private LDS staging region and private WMMA accumulators
// over an interleaved slice of the triple chunks; partials are combined in a
// fixed order (deterministic, no FP atomics).
// Transcendental diet per triple: cos(arccos(c)-s) folded to c*cos(s)+sqrt(1-c^2)*sin(s);
// pow(b, zeta) with b in [0,1] done as v_exp_f32(zeta * v_log_f32(b)).
__global__ __launch_bounds__(128)
void aev_wmma_kernel(const float* __restrict__ coords,
                     const float* __restrict__ EtaR,
                     const float* __restrict__ ShfR,
                     const float* __restrict__ EtaA,
                     const float* __restrict__ Zeta,
                     const float* __restrict__ ShfA,
                     const float* __restrict__ ShfZ,
                     const int*   __restrict__ species,
                     float* __restrict__ out)
{
    // per-wave staging: 32 triples x 32 features, stride 36 floats
    __shared__ __align__(16) float s_val[NWAVE * 32 * 36];
    __shared__ int            s_bin[NWAVE * 32];
    __shared__ __align__(16) float4 s_slot[ATOMS];  // {ux,uy,uz,d} per neighbor slot
    __shared__ float          s_fca[ATOMS];         // cutoff-cos at RCA per slot
    __shared__ float          s_fcr[ATOMS];         // cutoff-cos at RCR per slot
    __shared__ int            s_ssp[ATOMS];         // species per neighbor slot
    __shared__ unsigned short s_pair[NITER * 32];   // (slot_a<<6)|slot_b

    const int tid  = threadIdx.x;
    const int wave = tid >> 5;
    const int lane = tid & 31;
    const int m    = blockIdx.x / ATOMS;
    const int i    = blockIdx.x % ATOMS;

    const float cxi = coords[(m * ATOMS + i) * 3 + 0];
    const float cyi = coords[(m * ATOMS + i) * 3 + 1];
    const float czi = coords[(m * ATOMS + i) * 3 + 2];

    // ---- per-neighbor-slot geometry, computed once ----
    if (tid < NSLOT) {
        const int j = tid + (tid >= i);
        const float dx = coords[(m * ATOMS + j) * 3 + 0] - cxi;
        const float dy = coords[(m * ATOMS + j) * 3 + 1] - cyi;
        const float dz = coords[(m * ATOMS + j) * 3 + 2] - czi;
        const float d  = sqrtf(dx * dx + dy * dy + dz * dz);
        const float inv = 1.0f / d;
        s_slot[tid] = make_float4(dx * inv, dy * inv, dz * inv, d);
        s_fca[tid]  = 0.5f * __cosf(d * (PI_F / RCA_F)) + 0.5f;
        s_fcr[tid]  = 0.5f * __cosf(d * (PI_F / RCR_F)) + 0.5f;
        s_ssp[tid]  = species[m * ATOMS + j];
    }
    // ---- triple pair table over 47 slots: row per thread ----
    if (tid < NSLOT - 1) {
        const int a   = tid;
        const int off = 46 * a - (a * (a - 1)) / 2;
        for (int b = a + 1; b < NSLOT; ++b)
            s_pair[off + (b - a - 1)] = (unsigned short)((a << 6) | b);
    }
    if (tid < NITER * 32 - NTRI) s_pair[NTRI + tid] = 0;   // padding triples
    __syncthreads();

    const float etaR = EtaR[0], etaA = EtaA[0], zeta = Zeta[0];
    float shfR[16];
#pragma unroll
    for (int k = 0; k < 16; ++k) shfR[k] = ShfR[k];
    float shfA[4];
#pragma unroll
    for (int k = 0; k < 4; ++k) shfA[k] = ShfA[k];
    float csz[8], snz[8];                 // cos/sin of ShfZ, loop-invariant
#pragma unroll
    for (int k = 0; k < 8; ++k) __sincosf(ShfZ[k], &snz[k], &csz[k]);

    float* const sv  = s_val + wave * (32 * 36);
    int*   const sb  = s_bin + wave * 32;
    const int row  = lane & 15;          // WMMA M-row this lane owns
    const int koff = (lane >> 4) << 1;   // K sub-offset within a chunk: 0 or 2

    //------------------------------------------------------------------
    // Radial (waves 0 and 1, one 32-slot round each):
    //   D(16x16) += onehot(species_j)[16x4] x rt[4x16]
    //------------------------------------------------------------------
    v8f accR = {0.f, 0.f, 0.f, 0.f, 0.f, 0.f, 0.f, 0.f};
    if (wave < 2) {                       // wave-uniform branch, EXEC stays full
        const int  s     = wave * 32 + lane;
        const bool valid = (s < NSLOT);
        const int  sc    = valid ? s : 0;
        const float d    = s_slot[sc].w;
        const float w0   = (valid && d <= RCR_F) ? 0.25f * s_fcr[sc] : 0.f;
#pragma unroll
        for (int k = 0; k < 16; ++k) {
            const float q = d - shfR[k];
            sv[lane * 36 + k] = w0 * __expf(-etaR * q * q);
        }
        sb[lane] = valid ? s_ssp[sc] : 15;
        __builtin_amdgcn_wave_barrier();
#pragma unroll
        for (int c = 0; c < 8; ++c) {
            const int k0 = c * 4 + koff;
            const int b0 = sb[k0], b1 = sb[k0 + 1];
            v2f Aoh, Bv;
            Aoh.x = (b0 == row) ? 1.f : 0.f;
            Aoh.y = (b1 == row) ? 1.f : 0.f;
            Bv.x  = sv[k0 * 36 + row];
            Bv.y  = sv[(k0 + 1) * 36 + row];
            accR = __builtin_amdgcn_wmma_f32_16x16x4_f32(
                       false, Aoh, false, Bv, (short)0, accR, false, false);
        }
        __builtin_amdgcn_wave_barrier();
    }

    //------------------------------------------------------------------
    // Angular: each wave takes chunks it = wave, wave+4, ...
    // two 16x16 accumulators (feature cols 0-15 / 16-31)
    //------------------------------------------------------------------
    v8f acc0 = {0.f, 0.f, 0.f, 0.f, 0.f, 0.f, 0.f, 0.f};
    v8f acc1 = {0.f, 0.f, 0.f, 0.f, 0.f, 0.f, 0.f, 0.f};
    for (int it = wave; it < NITER; it += NWAVE) {
        const int t  = it * 32 + lane;
        const int pr = s_pair[t];
        const int a  = pr >> 6, b = pr & 63;      // neighbor slots (pad -> 0,0)
        const float4 r1 = s_slot[a];
        const float4 r2 = s_slot[b];
        const float d1 = r1.w, d2 = r2.w;
        const float mask = (t < NTRI && d1 <= RCA_F && d2 <= RCA_F) ? 1.f : 0.f;
        const float cosv = 0.95f * (r1.x * r2.x + r1.y * r2.y + r1.z * r2.z);
        const float sinv = sqrtf(fmaxf(1.f - cosv * cosv, 0.f));  // sin(arccos(cosv))
        const float dm   = 0.5f * (d1 + d2);
        const float h    = 2.f * mask * s_fca[a] * s_fca[b];
        v4f f1lo, f1hi;   // ((1+cos(ang-shfz))/2)^zeta, zeta*log2 -> exp2 on raw TRANS
#pragma unroll
        for (int sz = 0; sz < 4; ++sz) {
            const float cd = cosv * csz[sz] + sinv * snz[sz];
            f1lo[sz] = __builtin_amdgcn_exp2f(zeta * __builtin_amdgcn_logf(0.5f + 0.5f * cd));
        }
#pragma unroll
        for (int sz = 0; sz < 4; ++sz) {
            const float cd = cosv * csz[4 + sz] + sinv * snz[4 + sz];
            f1hi[sz] = __builtin_amdgcn_exp2f(zeta * __builtin_amdgcn_logf(0.5f + 0.5f * cd));
        }
#pragma unroll
        for (int sa = 0; sa < 4; ++sa) {
            const float q = dm - shfA[sa];
            const float g = h * __expf(-etaA * q * q);
            const v4f q0 = f1lo * g;     // packed muls
            const v4f q1 = f1hi * g;
            *(v4f*)&sv[lane * 36 + sa * 8 + 0] = q0;
            *(v4f*)&sv[lane * 36 + sa * 8 + 4] = q1;
        }
        const int sp1 = s_ssp[a], sp2 = s_ssp[b];
        const int lo = (sp1 < sp2) ? sp1 : sp2;
        const int hi = (sp1 < sp2) ? sp2 : sp1;
        sb[lane] = lo * (7 - lo) / 2 + hi;        // triu species-pair index 0..9
        __builtin_amdgcn_wave_barrier();
#pragma unroll
        for (int c = 0; c < 8; ++c) {
            const int k0 = c * 4 + koff;
            const int b0 = sb[k0], b1 = sb[k0 + 1];
            v2f Aoh, B0, B1;
            Aoh.x = (b0 == row) ? 1.f : 0.f;
            Aoh.y = (b1 == row) ? 1.f : 0.f;
            B0.x = sv[k0 * 36 + row];
            B0.y = sv[(k0 + 1) * 36 + row];
            B1.x = sv[k0 * 36 + 16 + row];
            B1.y = sv[(k0 + 1) * 36 + 16 + row];
            acc0 = __builtin_amdgcn_wmma_f32_16x16x4_f32(
                       false, Aoh, false, B0, (short)0, acc0, false, false);
            acc1 = __builtin_amdgcn_wmma_f32_16x16x4_f32(
                       false, Aoh, false, B1, (short)0, acc1, false, false);
        }
        __builtin_amdgcn_wave_barrier();
    }

    //------------------------------------------------------------------
    // Deterministic cross-wave reduction (fixed order w0 + w1 + w2 + w3)
    //------------------------------------------------------------------
    if (wave > 0) {
        float* p = s_val + wave * (32 * 36) + lane * 24;
#pragma unroll
        for (int r = 0; r < 8; ++r) {
            p[r]      = accR[r];
            p[8 + r]  = acc0[r];
            p[16 + r] = acc1[r];
        }
    }
    __syncthreads();
    if (wave == 0) {
#pragma unroll
        for (int w = 1; w < NWAVE; ++w) {
            const float* p = s_val + w * (32 * 36) + lane * 24;
#pragma unroll
            for (int r = 0; r < 8; ++r) {
                accR[r] += p[r];
                acc0[r] += p[8 + r];
                acc1[r] += p[16 + r];
            }
        }
        float* const orow = out + (size_t)(m * ATOMS + i) * FEAT;
        const int rb = (lane >> 4) * 8;
#pragma unroll
        for (int r = 0; r < 8; ++r) {
            const int sr = rb + r;
            if (sr < 4)  orow[sr * 16 + row] = accR[r];           // radial[sp*16+shfr]
            if (sr < 10) {
                orow[64 + sr * 32 + row]      = acc0[r];          // angular[pidx*32+..]
                orow[64 + sr * 32 + 16 + row] = acc1[r];
            }
        }
    }
}

extern "C" void kernel_launch(void* const* d_in, const int* in_sizes, int n_in,
                              void* d_out, int out_size, void* d_ws, size_t ws_size,
                              hipStream_t stream)
{
    const float* coords  = (const float*)d_in[0];
    const float* EtaR    = (const float*)d_in[1];
    const float* ShfR    = (const float*)d_in[2];
    const float* EtaA    = (const float*)d_in[3];
    const float* Zeta    = (const float*)d_in[4];
    const float* ShfA    = (const float*)d_in[5];
    const float* ShfZ    = (const float*)d_in[6];
    const int*   species = (const int*)d_in[7];
    const int M = in_sizes[0] / (ATOMS * 3);   // 32 for the reference setup

    aev_wmma_kernel<<<M * ATOMS, 128, 0, stream>>>(
        coords, EtaR, ShfR, EtaA, Zeta, ShfA, ShfZ, species, (float*)d_out);

    (void)n_in; (void)out_size; (void)d_ws; (void)ws_size;
}